// Embedding_34969623724740
// MI455X (gfx1250) — compile-verified
//
#include <hip/hip_runtime.h>

// Embedding gather: out[b, 0, 0, :] = table[idx1[b]], out[b, 0, 1, :] = table[idx2[b]]
// table: [VOCAB, 256] f32, idx: [B] i32, out: [B, 1, 2, 256] f32.
//
// Memory-bound (zero FLOPs). Strategy:
//  - one wave32 per 1KB output row; 2x B128 loads + 2x B128 NT stores per lane
//  - index scalarized via readfirstlane -> SGPR base addressing
//  - NT stores so the 256MB write-once output doesn't evict the ~100MB table
//    (which fits in MI455X's 192MB L2) -> HBM traffic ~356MB, ~15us floor.

typedef __attribute__((ext_vector_type(4))) float f32x4;

#define EMB 256           // floats per row
#define WAVES_PER_BLOCK 8 // 256 threads / wave32

__global__ __launch_bounds__(256) void embed_gather_f32(
    const int* __restrict__ idx1,
    const int* __restrict__ idx2,
    const float* __restrict__ table,
    float* __restrict__ out,
    int nrows)  // nrows = 2*B ; row r -> sample r>>1, slot r&1
{
    const int wave = (blockIdx.x * WAVES_PER_BLOCK) + (threadIdx.x >> 5);
    if (wave >= nrows) return;
    const int lane = threadIdx.x & 31;

    const int sample = wave >> 1;
    const int* __restrict__ ip = (wave & 1) ? idx2 : idx1;

    // All lanes read the same index; force it into an SGPR so the row base
    // is scalar and the loads become saddr-form global_load_b128.
    int row = ip[sample];
    row = __builtin_amdgcn_readfirstlane(row);

    const f32x4* __restrict__ src =
        (const f32x4*)(table + (size_t)row * EMB);
    f32x4* __restrict__ dst =
        (f32x4*)(out + (size_t)wave * EMB);

    // 256 floats = 64 float4; 32 lanes x 2 -> two fully-coalesced 128B/lane ops.
    f32x4 a = src[lane];        // RT load: keep table rows hot in 192MB L2
    f32x4 b = src[lane + 32];

    __builtin_nontemporal_store(a, dst + lane);        // NT: streaming output,
    __builtin_nontemporal_store(b, dst + lane + 32);   // don't pollute L2
}

extern "C" void kernel_launch(void* const* d_in, const int* in_sizes, int n_in,
                              void* d_out, int out_size, void* d_ws, size_t ws_size,
                              hipStream_t stream) {
    const int*   idx1  = (const int*)d_in[0];
    const int*   idx2  = (const int*)d_in[1];
    const float* table = (const float*)d_in[2];
    float*       out   = (float*)d_out;

    const int batch = in_sizes[0];   // 131072
    const int nrows = 2 * batch;     // one wave per 256-float row

    const int blocks = (nrows + WAVES_PER_BLOCK - 1) / WAVES_PER_BLOCK;
    embed_gather_f32<<<blocks, 256, 0, stream>>>(idx1, idx2, table, out, nrows);
}